// lossFunc_89876485636307
// MI455X (gfx1250) — compile-verified
//
#include <hip/hip_runtime.h>

// Problem constants from the reference
#define NUM_Q   4096
#define TSTEPS  199          // MAX_STEP - 1
#define BATCH_B 64
#define EPS_F   1e-8f

typedef __attribute__((ext_vector_type(2))) float v2f;
typedef __attribute__((ext_vector_type(4))) float v4f;
typedef __attribute__((ext_vector_type(8))) float v8f;

static constexpr int N4   = BATCH_B * TSTEPS * (NUM_Q / 4); // 13,041,664 float4 elements
static constexpr int NBLK = 4096;                            // partial-sum blocks
static constexpr int BLK  = 256;                             // 8 waves (wave32)

// Full-wave (32 lane) sum via V_WMMA_F32_16X16X4_F32.
// A layout (16x4 f32): lane L in [0,16): a[0]=A[L][0], a[1]=A[L][1];
//                      lane L in [16,32): a[0]=A[L-16][2], a[1]=A[L-16][3].
// With a = {acc, 0} and B = ones: D[m][n] = acc[m] + acc[m+16] for every n.
// D layout: VGPR r, lanes 0-15 -> row r; lanes 16-31 -> row r+8.
// Sum of the 8 D regs gives rows 0-7 total (lanes 0-15) / rows 8-15 total
// (lanes 16-31); one shfl_xor(16) completes the 32-lane sum in every lane.
// Requires EXEC == all ones (call only from fully converged code).
__device__ __forceinline__ float wave_sum_wmma(float acc) {
    v2f a; a[0] = acc;  a[1] = 0.0f;
    v2f b; b[0] = 1.0f; b[1] = 1.0f;
    v8f c = {};
    c = __builtin_amdgcn_wmma_f32_16x16x4_f32(
        /*neg_a=*/false, a, /*neg_b=*/false, b,
        /*c_mod=*/(short)0, c, /*reuse_a=*/false, /*reuse_b=*/false);
    float s = ((c[0] + c[1]) + (c[2] + c[3])) + ((c[4] + c[5]) + (c[6] + c[7]));
    s += __shfl_xor(s, 16, 32);
    return s;
}

// Pass 1: streaming BCE partial sums. Three coalesced b128 streams with
// non-temporal hints (626 MB read-once >> 192 MB L2), fast hardware log,
// per-block partial written to workspace.
__global__ void __launch_bounds__(BLK)
bce_partial_kernel(const float* __restrict__ pred,
                   const float* __restrict__ batch,
                   float* __restrict__ ws) {
    const v4f* __restrict__ pred4 = (const v4f*)pred;
    const v4f* __restrict__ bat4  = (const v4f*)batch;

    float acc = 0.0f;
    const int stride = gridDim.x * blockDim.x;
    for (int j = blockIdx.x * blockDim.x + threadIdx.x; j < N4; j += stride) {
        const int qv  = j & 1023;         // float4 index within the 4096-wide row
        const int row = j >> 10;          // b*199 + t
        const int b   = row / 199;        // compiler magic-multiply
        const int t   = row - b * 199;
        const int bb  = (b * 200 + t + 1) * 2048;   // batch row, float4 units

        const v4f p  = __builtin_nontemporal_load(&pred4[j]);
        const v4f mq = __builtin_nontemporal_load(&bat4[bb + qv]);        // mask_q
        const v4f mn = __builtin_nontemporal_load(&bat4[bb + 1024 + qv]); // mask_nq

        acc = fmaf(mq.x, __logf(p.x + EPS_F), acc);
        acc = fmaf(mn.x, __logf(1.0f - p.x + EPS_F), acc);
        acc = fmaf(mq.y, __logf(p.y + EPS_F), acc);
        acc = fmaf(mn.y, __logf(1.0f - p.y + EPS_F), acc);
        acc = fmaf(mq.z, __logf(p.z + EPS_F), acc);
        acc = fmaf(mn.z, __logf(1.0f - p.z + EPS_F), acc);
        acc = fmaf(mq.w, __logf(p.w + EPS_F), acc);
        acc = fmaf(mn.w, __logf(1.0f - p.w + EPS_F), acc);
    }

    // Converged here: WMMA-based 32-lane reduction, then 8-wave LDS combine.
    const float wsum = wave_sum_wmma(acc);

    __shared__ float lds[BLK / 32];
    const int lane = threadIdx.x & 31;
    const int wid  = threadIdx.x >> 5;
    if (lane == 0) lds[wid] = wsum;
    __syncthreads();
    if (threadIdx.x == 0) {
        float tsum = 0.0f;
        #pragma unroll
        for (int i = 0; i < BLK / 32; ++i) tsum += lds[i];
        ws[blockIdx.x] = tsum;
    }
}

// Pass 2: single block reduces NBLK partials; negates per the reference
// (loss = -sum). Deterministic: fixed tree, no atomics.
__global__ void __launch_bounds__(BLK)
bce_final_kernel(const float* __restrict__ ws, float* __restrict__ out) {
    float acc = 0.0f;
    for (int i = threadIdx.x; i < NBLK; i += BLK) acc += ws[i];

    const float wsum = wave_sum_wmma(acc);

    __shared__ float lds[BLK / 32];
    const int lane = threadIdx.x & 31;
    const int wid  = threadIdx.x >> 5;
    if (lane == 0) lds[wid] = wsum;
    __syncthreads();
    if (threadIdx.x == 0) {
        float tsum = 0.0f;
        #pragma unroll
        for (int i = 0; i < BLK / 32; ++i) tsum += lds[i];
        out[0] = -tsum;
    }
}

extern "C" void kernel_launch(void* const* d_in, const int* in_sizes, int n_in,
                              void* d_out, int out_size, void* d_ws, size_t ws_size,
                              hipStream_t stream) {
    (void)in_sizes; (void)n_in; (void)out_size; (void)ws_size;
    const float* pred  = (const float*)d_in[0];   // (64, 199, 4096) f32
    const float* batch = (const float*)d_in[1];   // (64, 200, 8192) f32
    float* out = (float*)d_out;                   // scalar f32
    float* ws  = (float*)d_ws;                    // NBLK partial sums (16 KB)

    bce_partial_kernel<<<NBLK, BLK, 0, stream>>>(pred, batch, ws);
    bce_final_kernel<<<1, BLK, 0, stream>>>(ws, out);
}